// bMomentumLIF_523986010148
// MI455X (gfx1250) — compile-verified
//
#include <hip/hip_runtime.h>

// momentum-LIF forward: B=64, T=64, F=8192, float32 in/out.
// HBM-streaming recurrence: 268 MB @ 23.3 TB/s ~= 11.5 us floor.
// Grid is work-limited (4096 waves), so latency tolerance must come from
// per-wave pipeline depth: DEPTH=8 outstanding B128 loads/wave = 4KB/wave
// in flight (~16MB device-wide ~= BW x HBM-latency product).

#define T_STEPS 64
#define F_DIM   8192
#define FV      (F_DIM / 4)   // 2048 vec4 per (b,t) row
#define BLOCK   256
#define DEPTH   8             // register pipeline depth (power of 2)

// native clang vector (NOT HIP_vector_type) so nontemporal builtins accept it
typedef float v4f __attribute__((ext_vector_type(4)));

__global__ __launch_bounds__(BLOCK, 8)
void lif_fwd_kernel(const float* __restrict__ x,
                    const float* __restrict__ momentum,
                    const float* __restrict__ lamb,
                    const float* __restrict__ thresholds,
                    float* __restrict__ out,
                    int total_vec) {
    const int tid = blockIdx.x * BLOCK + threadIdx.x;
    if (tid >= total_vec) return;

    // uniform scalars (1-element device arrays)
    const float mt  = momentum[0];
    const float lb  = lamb[0];
    const float th  = thresholds[0];
    const float omt = 1.0f - mt;
    const float olb = 1.0f - lb;

    const int b  = tid / FV;
    const int fv = tid - b * FV;

    const v4f* __restrict__ xin  = reinterpret_cast<const v4f*>(x);
    v4f*       __restrict__ sout = reinterpret_cast<v4f*>(out);

    const size_t base = (size_t)b * T_STEPS * FV + (size_t)fv;

    // recurrence state as scalars (VGPR-resident)
    float u0 = 0.f, u1 = 0.f, u2 = 0.f, u3 = 0.f;
    float m0 = 0.f, m1 = 0.f, m2 = 0.f, m3 = 0.f;

    auto step = [&](float& u_carry, float& m, float xi) -> float {
        const float ul = u_carry;
        const float u  = fmaf(ul, 0.5f, xi);            // u_last / TAU + x
        const float s  = (u >= th) ? 1.0f : 0.0f;       // heaviside spike
        m = fmaf(mt, m, omt * (u - ul));                // momentum update
        u_carry = fmaf(u, lb, m * olb) * (1.0f - s);    // leak + reset
        return s;
    };

    // ---- prologue: fill DEPTH-slot register pipeline ----
    v4f xb[DEPTH];
#pragma unroll
    for (int i = 0; i < DEPTH; ++i)
        xb[i] = __builtin_nontemporal_load(&xin[base + (size_t)i * FV]);

    // ---- main loop: compute t, refill slot with t+DEPTH ----
#pragma unroll DEPTH
    for (int t = 0; t < T_STEPS - DEPTH; ++t) {
        // long-range L2 warm for the 32KB-strided T-walk (global_prefetch_b8)
        if (t + 2 * DEPTH < T_STEPS)
            __builtin_prefetch(&xin[base + (size_t)(t + 2 * DEPTH) * FV], 0, 0);

        const v4f xi = xb[t & (DEPTH - 1)];

        v4f s;
        s.x = step(u0, m0, xi.x);
        s.y = step(u1, m1, xi.y);
        s.z = step(u2, m2, xi.z);
        s.w = step(u3, m3, xi.w);

        // refill this slot; consumed DEPTH iterations from now
        xb[t & (DEPTH - 1)] =
            __builtin_nontemporal_load(&xin[base + (size_t)(t + DEPTH) * FV]);

        // streamed write-once output: NT store (don't thrash 192MB L2)
        __builtin_nontemporal_store(s, &sout[base + (size_t)t * FV]);
    }

    // ---- epilogue: drain pipeline, no more loads ----
#pragma unroll
    for (int t = T_STEPS - DEPTH; t < T_STEPS; ++t) {
        const v4f xi = xb[t & (DEPTH - 1)];

        v4f s;
        s.x = step(u0, m0, xi.x);
        s.y = step(u1, m1, xi.y);
        s.z = step(u2, m2, xi.z);
        s.w = step(u3, m3, xi.w);

        __builtin_nontemporal_store(s, &sout[base + (size_t)t * FV]);
    }
}

extern "C" void kernel_launch(void* const* d_in, const int* in_sizes, int n_in,
                              void* d_out, int out_size, void* d_ws, size_t ws_size,
                              hipStream_t stream) {
    const float* x          = (const float*)d_in[0];
    const float* momentum   = (const float*)d_in[1];
    const float* lamb       = (const float*)d_in[2];
    const float* thresholds = (const float*)d_in[3];
    float*       out        = (float*)d_out;

    // derive B from input size (T, F fixed by the reference)
    const int B = in_sizes[0] / (T_STEPS * F_DIM);
    const int total_vec = B * FV;                 // 131072 for B=64
    const int blocks = (total_vec + BLOCK - 1) / BLOCK;

    hipLaunchKernelGGL(lif_fwd_kernel, dim3(blocks), dim3(BLOCK), 0, stream,
                       x, momentum, lamb, thresholds, out, total_vec);
}